// LASLNNet_7791070675391
// MI455X (gfx1250) — compile-verified
//
#include <hip/hip_runtime.h>
#include <hip/hip_bf16.h>

typedef __attribute__((ext_vector_type(2))) float v2f;
typedef __attribute__((ext_vector_type(8))) float v8f;

// ---------------------------------------------------------------------------
// fp32 WMMA: D(16x16,f32) = A(16x4,f32) * B(4x16,f32) + C
// ---------------------------------------------------------------------------
static __device__ __forceinline__ v8f wmma4(v2f a, v2f b, v8f c) {
  return __builtin_amdgcn_wmma_f32_16x16x4_f32(
      /*neg_a=*/false, a, /*neg_b=*/false, b,
      /*c_mod=*/(short)0, c, /*reuse_a=*/false, /*reuse_b=*/false);
}

// ---------------------------------------------------------------------------
// Per-layer static configuration (batch B = 4 everywhere)
// ---------------------------------------------------------------------------
template <int L> struct Cfg {};
template <> struct Cfg<1> { static constexpr int Cin=1,   Cout=32,  KK=3, S=2, PADV=0, DIN=20, DOUT=9; };
template <> struct Cfg<2> { static constexpr int Cin=32,  Cout=64,  KK=3, S=1, PADV=1, DIN=9,  DOUT=9; };
template <> struct Cfg<3> { static constexpr int Cin=64,  Cout=128, KK=1, S=1, PADV=0, DIN=9,  DOUT=9; };
template <> struct Cfg<4> { static constexpr int Cin=128, Cout=128, KK=1, S=1, PADV=0, DIN=9,  DOUT=9; };
template <> struct Cfg<5> { static constexpr int Cin=128, Cout=64,  KK=1, S=2, PADV=0, DIN=9,  DOUT=5; };

// Decoded output column (one lane's N position)
struct Col {
  int b;
  int o1, o2, o3, o4;
  int pOut;       // flattened output spatial index
  int colBase;    // (b*Cin)*DIN^4 + base input spatial offset (1x1 layers)
  bool valid;
};

template <int L>
static __device__ __forceinline__ Col decodeCol(int n) {
  using C = Cfg<L>;
  constexpr int P   = C::DOUT * C::DOUT * C::DOUT * C::DOUT;
  constexpr int PIN = C::DIN * C::DIN * C::DIN * C::DIN;
  Col c;
  c.valid = n < 4 * P;
  const int nc = c.valid ? n : 4 * P - 1;
  c.b = nc / P;
  int p = nc - c.b * P;
  c.pOut = p;
  c.o1 = p / (C::DOUT * C::DOUT * C::DOUT); p -= c.o1 * C::DOUT * C::DOUT * C::DOUT;
  c.o2 = p / (C::DOUT * C::DOUT);           p -= c.o2 * C::DOUT * C::DOUT;
  c.o3 = p / C::DOUT;
  c.o4 = p - c.o3 * C::DOUT;
  const int i1 = c.o1 * C::S, i2 = c.o2 * C::S, i3 = c.o3 * C::S, i4 = c.o4 * C::S;
  c.colBase = c.b * C::Cin * PIN +
              ((i1 * C::DIN + i2) * C::DIN + i3) * C::DIN + i4;
  return c;
}

// Generic im2col offset (layer 1): reduction index t -> input offset
template <int L>
static __device__ __forceinline__ int boffGeneric(const Col& c, int t) {
  using C = Cfg<L>;
  int ci = 0, j1 = 0, j2 = 0, j3 = 0, j4 = 0;
  if constexpr (C::KK == 3) {
    ci = t / 81; int r = t - ci * 81;
    j1 = r / 27; r -= j1 * 27;
    j2 = r / 9;  r -= j2 * 9;
    j3 = r / 3;  j4 = r - j3 * 3;
  } else {
    ci = t;
  }
  const int i1 = c.o1 * C::S + j1;
  const int i2 = c.o2 * C::S + j2;
  const int i3 = c.o3 * C::S + j3;
  const int i4 = c.o4 * C::S + j4;
  return (((((c.b * C::Cin + ci) * C::DIN + i1) * C::DIN + i2) * C::DIN + i3) * C::DIN) + i4;
}

// Build WMMA operands from two interleaved complex loads (K-pair k, k+1):
//   p0 = (re[k], im[k]), p1 = (re[k+1], im[k+1])
static __device__ __forceinline__ void splitPair(v2f p0, v2f p1,
                                                 v2f& re, v2f& im, v2f& nim) {
  re.x = p0.x; re.y = p1.x;
  im.x = p0.y; im.y = p1.y;
  nim = -im;
}

// ---------------------------------------------------------------------------
// Complex conv as implicit GEMM on the WMMA pipe.
// Activations and weights are interleaved complex (float2 per element), so
// each operand K element is a single global_load_b64.
// One wave = 16(M) x 32(N) tile (two 16x16 sub-tiles sharing A operands).
//   accR += Wr*Xr + Wi*(-Xi)     accI += Wr*Xi + Wi*Xr
// ---------------------------------------------------------------------------
template <int L>
__global__ void __launch_bounds__(256) cconv_wmma(
    const v2f* __restrict__ xc,   // interleaved input activations
    const v2f* __restrict__ wc,   // interleaved weights [co][K]
    const float* __restrict__ br, const float* __restrict__ bi,
    v2f* __restrict__ yc) {       // interleaved output activations
  using C = Cfg<L>;
  constexpr int KTOT = C::Cin * C::KK * C::KK * C::KK * C::KK;
  constexpr int P    = C::DOUT * C::DOUT * C::DOUT * C::DOUT;
  constexpr int PIN  = C::DIN * C::DIN * C::DIN * C::DIN;
  constexpr int NTOT = 4 * P;
  constexpr int MT   = C::Cout / 16;
  constexpr int NT2  = (NTOT + 31) / 32;

  const int wave = blockIdx.x * (blockDim.x >> 5) + (threadIdx.x >> 5);
  if (wave >= MT * NT2) return;  // wave-uniform exit

  const int lane = threadIdx.x & 31;
  const int half = lane >> 4;    // 0: K {0,1} / rows 0..7   1: K {2,3} / rows 8..15
  const int l15  = lane & 15;

  const int mTile = wave % MT;
  const int nPair = wave / MT;

  const Col cA = decodeCol<L>(nPair * 32 + l15);
  const Col cB = decodeCol<L>(nPair * 32 + 16 + l15);

  const int mRowA = mTile * 16 + l15;          // Cout is a multiple of 16
  const v2f* __restrict__ wcRow = wc + mRowA * KTOT;

  v8f accRA = {}, accIA = {}, accRB = {}, accIB = {};
  const v2f vzero = {};

  if constexpr (C::KK == 3 && C::Cin > 1) {
    // ---------------- Layer-2 path: j outer, ci inner ----------------
    constexpr int DD = C::DIN;
#pragma unroll 1
    for (int j = 0; j < 81; ++j) {
      const int j1 = j / 27; int r = j - j1 * 27;
      const int j2 = r / 9;  r -= j2 * 9;
      const int j3 = r / 3;  const int j4 = r - j3 * 3;

      const int iA1 = cA.o1 + j1 - 1, iA2 = cA.o2 + j2 - 1,
                iA3 = cA.o3 + j3 - 1, iA4 = cA.o4 + j4 - 1;
      const int iB1 = cB.o1 + j1 - 1, iB2 = cB.o2 + j2 - 1,
                iB3 = cB.o3 + j3 - 1, iB4 = cB.o4 + j4 - 1;
      const bool vA = cA.valid &
          ((unsigned)iA1 < (unsigned)DD) & ((unsigned)iA2 < (unsigned)DD) &
          ((unsigned)iA3 < (unsigned)DD) & ((unsigned)iA4 < (unsigned)DD);
      const bool vB = cB.valid &
          ((unsigned)iB1 < (unsigned)DD) & ((unsigned)iB2 < (unsigned)DD) &
          ((unsigned)iB3 < (unsigned)DD) & ((unsigned)iB4 < (unsigned)DD);
      const int baseA = cA.b * C::Cin * PIN + ((iA1 * DD + iA2) * DD + iA3) * DD + iA4;
      const int baseB = cB.b * C::Cin * PIN + ((iB1 * DD + iB2) * DD + iB3) * DD + iB4;

#pragma unroll
      for (int cb = 0; cb < C::Cin; cb += 4) {
        const int ciL = cb + 2 * half;
        // A: one b64 per K element (wr, wi interleaved), stride 81 along ci
        const v2f w0 = wcRow[ciL * 81 + j];
        const v2f w1 = wcRow[(ciL + 1) * 81 + j];
        v2f aR, aI, aNunused;
        aR.x = w0.x; aR.y = w1.x;
        aI.x = w0.y; aI.y = w1.y;
        (void)aNunused;
        // B: one b64 per (t, n), stride PIN along ci
        v2f pA0 = vzero, pA1 = vzero, pB0 = vzero, pB1 = vzero;
        if (vA) { const int oA = baseA + ciL * PIN; pA0 = xc[oA]; pA1 = xc[oA + PIN]; }
        if (vB) { const int oB = baseB + ciL * PIN; pB0 = xc[oB]; pB1 = xc[oB + PIN]; }
        v2f bRA, bIA, bNA, bRB, bIB, bNB;
        splitPair(pA0, pA1, bRA, bIA, bNA);
        splitPair(pB0, pB1, bRB, bIB, bNB);
        accRA = wmma4(aR, bRA, accRA);  accRA = wmma4(aI, bNA, accRA);
        accIA = wmma4(aR, bIA, accIA);  accIA = wmma4(aI, bRA, accIA);
        accRB = wmma4(aR, bRB, accRB);  accRB = wmma4(aI, bNB, accRB);
        accIB = wmma4(aR, bIB, accIB);  accIB = wmma4(aI, bRB, accIB);
      }
    }
  } else if constexpr (C::KK == 1) {
    // ---------------- 1x1 layers: linear offsets along ci ----------------
#pragma unroll 4
    for (int cb = 0; cb < C::Cin; cb += 4) {
      const int t0 = cb + 2 * half;
      const v2f w0 = wcRow[t0];
      const v2f w1 = wcRow[t0 + 1];
      v2f aR, aI;
      aR.x = w0.x; aR.y = w1.x;
      aI.x = w0.y; aI.y = w1.y;
      v2f pA0 = vzero, pA1 = vzero, pB0 = vzero, pB1 = vzero;
      if (cA.valid) { const int oA = cA.colBase + t0 * PIN; pA0 = xc[oA]; pA1 = xc[oA + PIN]; }
      if (cB.valid) { const int oB = cB.colBase + t0 * PIN; pB0 = xc[oB]; pB1 = xc[oB + PIN]; }
      v2f bRA, bIA, bNA, bRB, bIB, bNB;
      splitPair(pA0, pA1, bRA, bIA, bNA);
      splitPair(pB0, pB1, bRB, bIB, bNB);
      accRA = wmma4(aR, bRA, accRA);  accRA = wmma4(aI, bNA, accRA);
      accIA = wmma4(aR, bIA, accIA);  accIA = wmma4(aI, bRA, accIA);
      accRB = wmma4(aR, bRB, accRB);  accRB = wmma4(aI, bNB, accRB);
      accIB = wmma4(aR, bIB, accIB);  accIB = wmma4(aI, bRB, accIB);
    }
  } else {
    // ---------------- Layer 1 (Cin=1, K=81, tail-guarded) ----------------
#pragma unroll 1
    for (int kb = 0; kb < KTOT; kb += 4) {
      const int t0 = kb + 2 * half;
      const int t1 = t0 + 1;
      const bool tv0 = t0 < KTOT, tv1 = t1 < KTOT;
      const v2f w0 = tv0 ? wcRow[t0] : vzero;
      const v2f w1 = tv1 ? wcRow[t1] : vzero;
      v2f aR, aI;
      aR.x = w0.x; aR.y = w1.x;
      aI.x = w0.y; aI.y = w1.y;
      const int oA0 = boffGeneric<L>(cA, tv0 ? t0 : 0);
      const int oA1 = boffGeneric<L>(cA, tv1 ? t1 : 0);
      const int oB0 = boffGeneric<L>(cB, tv0 ? t0 : 0);
      const int oB1 = boffGeneric<L>(cB, tv1 ? t1 : 0);
      v2f pA0 = vzero, pA1 = vzero, pB0 = vzero, pB1 = vzero;
      if (cA.valid & tv0) pA0 = xc[oA0];
      if (cA.valid & tv1) pA1 = xc[oA1];
      if (cB.valid & tv0) pB0 = xc[oB0];
      if (cB.valid & tv1) pB1 = xc[oB1];
      v2f bRA, bIA, bNA, bRB, bIB, bNB;
      splitPair(pA0, pA1, bRA, bIA, bNA);
      splitPair(pB0, pB1, bRB, bIB, bNB);
      accRA = wmma4(aR, bRA, accRA);  accRA = wmma4(aI, bNA, accRA);
      accIA = wmma4(aR, bIA, accIA);  accIA = wmma4(aI, bRA, accIA);
      accRB = wmma4(aR, bRB, accRB);  accRB = wmma4(aI, bNB, accRB);
      accIB = wmma4(aR, bIB, accIB);  accIB = wmma4(aI, bRB, accIB);
    }
  }

  // ------- Epilogue: bias + ReLU, one b64 store per output element -------
  const int rowBase = mTile * 16 + half * 8;
#pragma unroll
  for (int rI = 0; rI < 8; ++rI) {
    const int row = rowBase + rI;
    const float brv = br[row];
    const float biv = bi[row];
    if (cA.valid) {
      v2f o;
      float vr = accRA[rI] + brv; o.x = vr > 0.f ? vr : 0.f;
      float vi = accIA[rI] + biv; o.y = vi > 0.f ? vi : 0.f;
      yc[(cA.b * C::Cout + row) * P + cA.pOut] = o;
    }
    if (cB.valid) {
      v2f o;
      float vr = accRB[rI] + brv; o.x = vr > 0.f ? vr : 0.f;
      float vi = accIB[rI] + biv; o.y = vi > 0.f ? vi : 0.f;
      yc[(cB.b * C::Cout + row) * P + cB.pOut] = o;
    }
  }
}

// ---------------------------------------------------------------------------
// Interleave two real planes into complex float2 storage.
// ---------------------------------------------------------------------------
__global__ void __launch_bounds__(256) interleave2(
    const float* __restrict__ a, const float* __restrict__ b,
    v2f* __restrict__ o, int n) {
  const int i = blockIdx.x * blockDim.x + threadIdx.x;
  if (i < n) {
    v2f v; v.x = a[i]; v.y = b[i];
    o[i] = v;
  }
}

// ---------------------------------------------------------------------------
// Final FC: (4,64,625) . fcw(625) -> 512 outputs (yr | yi). act5 already ReLU'd.
// ---------------------------------------------------------------------------
__global__ void __launch_bounds__(256) fc_kernel(
    const v2f* __restrict__ a5c,
    const float* __restrict__ fcw, const float* __restrict__ fcb,
    float* __restrict__ out) {
  const int tid = blockIdx.x * blockDim.x + threadIdx.x;
  if (tid >= 256) return;
  const v2f* __restrict__ pc = a5c + tid * 625;
  float sr = 0.0f, si = 0.0f;
  for (int p = 0; p < 625; ++p) {
    const float w = fcw[p];
    const v2f v = pc[p];
    sr += v.x * w;
    si += v.y * w;
  }
  out[tid]       = sr + fcb[0];  // yr
  out[256 + tid] = si;           // yi
}

// ---------------------------------------------------------------------------
// Host launch
// ---------------------------------------------------------------------------
extern "C" void kernel_launch(void* const* d_in, const int* in_sizes, int n_in,
                              void* d_out, int out_size, void* d_ws, size_t ws_size,
                              hipStream_t stream) {
  const float* xr  = (const float*)d_in[0];
  const float* xi  = (const float*)d_in[1];
  const float* w1r = (const float*)d_in[2];  const float* w1i = (const float*)d_in[3];
  const float* b1r = (const float*)d_in[4];  const float* b1i = (const float*)d_in[5];
  const float* w2r = (const float*)d_in[6];  const float* w2i = (const float*)d_in[7];
  const float* b2r = (const float*)d_in[8];  const float* b2i = (const float*)d_in[9];
  const float* w3r = (const float*)d_in[10]; const float* w3i = (const float*)d_in[11];
  const float* b3r = (const float*)d_in[12]; const float* b3i = (const float*)d_in[13];
  const float* w4r = (const float*)d_in[14]; const float* w4i = (const float*)d_in[15];
  const float* b4r = (const float*)d_in[16]; const float* b4i = (const float*)d_in[17];
  const float* w5r = (const float*)d_in[18]; const float* w5i = (const float*)d_in[19];
  const float* b5r = (const float*)d_in[20]; const float* b5i = (const float*)d_in[21];
  const float* fcw = (const float*)d_in[22]; const float* fcb = (const float*)d_in[23];
  float* out = (float*)d_out;

  // Workspace layout (v2f units): interleaved x, weights, activations 1..5.
  const size_t NX  = (size_t)4 * 1 * 160000;   // 640,000
  const size_t NW1 = (size_t)32  * 81;         // 2,592
  const size_t NW2 = (size_t)64  * 2592;       // 165,888
  const size_t NW3 = (size_t)128 * 64;
  const size_t NW4 = (size_t)128 * 128;
  const size_t NW5 = (size_t)64  * 128;
  const size_t S1  = (size_t)4 * 32  * 6561;
  const size_t S2  = (size_t)4 * 64  * 6561;
  const size_t S3  = (size_t)4 * 128 * 6561;
  const size_t S5  = (size_t)4 * 64  * 625;

  v2f* ws  = (v2f*)d_ws;
  v2f* xcI = ws;
  v2f* w1c = xcI + NX;
  v2f* w2c = w1c + NW1;
  v2f* w3c = w2c + NW2;
  v2f* w4c = w3c + NW3;
  v2f* w5c = w4c + NW4;
  v2f* a1  = w5c + NW5;
  v2f* a2  = a1 + S1;
  v2f* a3  = a2 + S2;
  v2f* a4  = a3 + S3;
  v2f* a5  = a4 + S3;

  // --- prep: interleave input + weights into complex storage ---
  auto ilBlocks = [](size_t n) { return (int)((n + 255) / 256); };
  interleave2<<<ilBlocks(NX),  256, 0, stream>>>(xr,  xi,  xcI, (int)NX);
  interleave2<<<ilBlocks(NW1), 256, 0, stream>>>(w1r, w1i, w1c, (int)NW1);
  interleave2<<<ilBlocks(NW2), 256, 0, stream>>>(w2r, w2i, w2c, (int)NW2);
  interleave2<<<ilBlocks(NW3), 256, 0, stream>>>(w3r, w3i, w3c, (int)NW3);
  interleave2<<<ilBlocks(NW4), 256, 0, stream>>>(w4r, w4i, w4c, (int)NW4);
  interleave2<<<ilBlocks(NW5), 256, 0, stream>>>(w5r, w5i, w5c, (int)NW5);

  // blocks = ceil(mTiles * ceil(N/32) / 8 waves-per-block)
  auto blocksFor = [](int mTiles, int ntot) {
    const int tiles = mTiles * ((ntot + 31) / 32);
    return (tiles + 7) / 8;
  };
  const int N14 = 4 * 6561;
  const int N5  = 4 * 625;

  cconv_wmma<1><<<blocksFor(2, N14), 256, 0, stream>>>(xcI, w1c, b1r, b1i, a1);
  cconv_wmma<2><<<blocksFor(4, N14), 256, 0, stream>>>(a1,  w2c, b2r, b2i, a2);
  cconv_wmma<3><<<blocksFor(8, N14), 256, 0, stream>>>(a2,  w3c, b3r, b3i, a3);
  cconv_wmma<4><<<blocksFor(8, N14), 256, 0, stream>>>(a3,  w4c, b4r, b4i, a4);
  cconv_wmma<5><<<blocksFor(4, N5),  256, 0, stream>>>(a4,  w5c, b5r, b5i, a5);
  fc_kernel<<<1, 256, 0, stream>>>(a5, fcw, fcb, out);
}